// ScaledDotProductAttention_25847113187552
// MI455X (gfx1250) — compile-verified
//
#include <hip/hip_runtime.h>
#include <cstdint>

typedef float v2f __attribute__((ext_vector_type(2)));
typedef float v4f __attribute__((ext_vector_type(4)));
typedef float v8f __attribute__((ext_vector_type(8)));

#define BH 128
#define S 1024
#define D 64
#define INV_SCALE 0.044194173824159216f  // 1/sqrt(512)
#define MASK_FILL 1e-9f

// One block = 4 waves (128 threads) = one 16-row Q tile.
// LDS holds the 16x1024 fp32 score/prob tile (64 KB).
__global__ __launch_bounds__(128) void sdpa_wmma_f32_kernel(
    const float* __restrict__ Q,
    const float* __restrict__ K,
    const float* __restrict__ V,
    const unsigned char* __restrict__ Mask,  // jax bool -> 1 byte/elem
    float* __restrict__ Out,                 // [BH,S,D]
    float* __restrict__ Attn)                // [BH,S,S]
{
    __shared__ float smem[16 * S];  // 65536 bytes

    const int bh   = blockIdx.x >> 6;
    const int q0   = (blockIdx.x & 63) << 4;
    const int lane = threadIdx.x & 31;
    const int wave = threadIdx.x >> 5;

    const int mrow  = lane & 15;          // M (A) / N (B,C,D) index within 16
    const int koff  = (lane >> 4) << 1;   // 0 or 2: K-pair offset in chunk of 4
    const int rbase = (lane >> 4) << 3;   // C/D: VGPR r -> row r + rbase

    const float* q_base = Q + ((size_t)bh * S + q0) * D;
    const float* k_base = K + (size_t)bh * S * D;
    const float* v_base = V + (size_t)bh * S * D;
    const unsigned char* m_base = Mask + ((size_t)bh * S + q0) * S;
    float* attn_base = Attn + ((size_t)bh * S + q0) * S;
    float* out_base  = Out  + ((size_t)bh * S + q0) * D;

    // ---------------- Pass 1: scores = (Q K^T) * inv_scale -> LDS -------
    // Preload A fragments (Q tile): 16 chunks of K=4, 2 floats/lane each.
    v2f a_frag[16];
#pragma unroll
    for (int c = 0; c < 16; ++c)
        a_frag[c] = *(const v2f*)(q_base + mrow * D + 4 * c + koff);

    // Each wave computes 16 of the 64 16x16 score blocks.
    for (int t = wave * 16; t < wave * 16 + 16; ++t) {
        const float* kt = k_base + (size_t)t * 16 * D;  // 16 K-rows
        v8f acc = {};
#pragma unroll
        for (int c = 0; c < 16; ++c) {
            // B = K^T chunk: lane holds K[n=mrow][4c+koff .. +1]
            v2f b = *(const v2f*)(kt + mrow * D + 4 * c + koff);
            acc = __builtin_amdgcn_wmma_f32_16x16x4_f32(
                false, a_frag[c], false, b, (short)0, acc, false, false);
        }
#pragma unroll
        for (int r = 0; r < 8; ++r)
            smem[(rbase + r) * S + t * 16 + mrow] = acc[r] * INV_SCALE;
    }
    __syncthreads();

    // ---------------- Pass 2: masked softmax per row --------------------
    // Whole wave cooperates on one row; 4 rows per wave.
    for (int m = wave * 4; m < wave * 4 + 4; ++m) {
        float* srow = smem + m * S;
        const unsigned char* mask_row = m_base + (size_t)m * S;

        // scan 1: apply mask fill (1e-9 replaces scaled score), row max
        float mx = -3.0e38f;
#pragma unroll
        for (int k = 0; k < 8; ++k) {
            const int col = 4 * lane + 128 * k;
            v4f s = *(v4f*)(srow + col);
            // mask is read exactly once across the kernel: non-temporal
            const uint32_t mb =
                __builtin_nontemporal_load((const uint32_t*)(mask_row + col));
            s.x = (mb & 0x000000ffu) ? MASK_FILL : s.x;
            s.y = (mb & 0x0000ff00u) ? MASK_FILL : s.y;
            s.z = (mb & 0x00ff0000u) ? MASK_FILL : s.z;
            s.w = (mb & 0xff000000u) ? MASK_FILL : s.w;
            *(v4f*)(srow + col) = s;
            mx = fmaxf(mx, fmaxf(fmaxf(s.x, s.y), fmaxf(s.z, s.w)));
        }
#pragma unroll
        for (int off = 16; off > 0; off >>= 1)
            mx = fmaxf(mx, __shfl_xor(mx, off, 32));

        // scan 2: exponentiate, row sum
        float sum = 0.0f;
#pragma unroll
        for (int k = 0; k < 8; ++k) {
            const int col = 4 * lane + 128 * k;
            v4f s = *(v4f*)(srow + col);
            s.x = __expf(s.x - mx);
            s.y = __expf(s.y - mx);
            s.z = __expf(s.z - mx);
            s.w = __expf(s.w - mx);
            *(v4f*)(srow + col) = s;
            sum += (s.x + s.y) + (s.z + s.w);
        }
#pragma unroll
        for (int off = 16; off > 0; off >>= 1)
            sum += __shfl_xor(sum, off, 32);
        const float inv = 1.0f / sum;

        // scan 3: normalize; keep P in LDS, stream P to global.
        // attn is a 512 MB write-once stream: non-temporal stores keep the
        // 192 MB L2 reserved for the reused Q/K/V working set.
#pragma unroll
        for (int k = 0; k < 8; ++k) {
            const int col = 4 * lane + 128 * k;
            v4f s = *(v4f*)(srow + col);
            s.x *= inv; s.y *= inv; s.z *= inv; s.w *= inv;
            *(v4f*)(srow + col) = s;
            __builtin_nontemporal_store(
                s, (v4f*)(attn_base + (size_t)m * S + col));
        }
    }
    __syncthreads();

    // ---------------- Pass 3: out = P @ V (one n-tile per wave) ---------
    {
        const int n0 = wave * 16;
        v8f acc = {};
        for (int kc = 0; kc < 256; ++kc) {
            const int kk = 4 * kc + koff;
            // A = P chunk from LDS: lane holds P[m=mrow][kk .. kk+1]
            v2f a = *(const v2f*)(smem + mrow * S + kk);
            // B = V chunk: lane holds V[kk][n0+mrow], V[kk+1][n0+mrow]
            v2f b;
            b.x = v_base[(size_t)kk * D + n0 + mrow];
            b.y = v_base[(size_t)(kk + 1) * D + n0 + mrow];
            acc = __builtin_amdgcn_wmma_f32_16x16x4_f32(
                false, a, false, b, (short)0, acc, false, false);
        }
#pragma unroll
        for (int r = 0; r < 8; ++r)
            out_base[(size_t)(rbase + r) * D + n0 + mrow] = acc[r];
    }
}

extern "C" void kernel_launch(void* const* d_in, const int* in_sizes, int n_in,
                              void* d_out, int out_size, void* d_ws, size_t ws_size,
                              hipStream_t stream) {
    (void)in_sizes; (void)n_in; (void)out_size; (void)d_ws; (void)ws_size;
    const float* q = (const float*)d_in[0];
    const float* k = (const float*)d_in[1];
    const float* v = (const float*)d_in[2];
    const unsigned char* mask = (const unsigned char*)d_in[3];

    float* out  = (float*)d_out;                        // [BH,S,D] first
    float* attn = (float*)d_out + (size_t)BH * S * D;   // then [BH,S,S]

    dim3 grid(BH * (S / 16));   // 8192 blocks
    dim3 block(128);            // 4 waves: 16 q-rows per block
    hipLaunchKernelGGL(sdpa_wmma_f32_kernel, grid, block, 0, stream,
                       q, k, v, mask, out, attn);
}